// DATSwinDAttention_32375463477461
// MI455X (gfx1250) — compile-verified
//
#include <hip/hip_runtime.h>
#include <hip/hip_bf16.h>

typedef float v2f __attribute__((ext_vector_type(2)));
typedef float v8f __attribute__((ext_vector_type(8)));

#define CDIM 192
#define NTOK 64
#define NSAMP 16
#define NHEAD 6
#define NGRP 3
#define GCH 64
#define HCH 32

__device__ __forceinline__ v8f wmma_f32(v2f a, v2f b, v8f c) {
  // V_WMMA_F32_16X16X4_F32 : D(16x16,f32) = A(16x4,f32) x B(4x16,f32) + C
  return __builtin_amdgcn_wmma_f32_16x16x4_f32(false, a, false, b, (short)0, c, false, false);
}

// ---------------------------------------------------------------- q projection
// q[b][c][n] = sum_k wq[c][k] * x[b][n][k] + bq[c]
// 12 waves / block: wave owns one 16-row M tile, 4 N tiles share the A operand.
__global__ __launch_bounds__(384) void k_qproj(const float* __restrict__ x,
                                               const float* __restrict__ wq,
                                               const float* __restrict__ bq,
                                               float* __restrict__ q) {
  __shared__ float lx[NTOK * 196];  // [n][c], padded stride (196 % 64 == 4)
  int b = blockIdx.x;
  const float* xw = x + (size_t)b * NTOK * CDIM;
  for (int i = threadIdx.x; i < NTOK * CDIM; i += 384) {
    int n = i / CDIM, c = i % CDIM;
    lx[n * 196 + c] = xw[i];
  }
  __syncthreads();
  int wave = threadIdx.x >> 5, lane = threadIdx.x & 31;
  int half = lane >> 4, lm = lane & 15;
  float* qb = q + (size_t)b * CDIM * NTOK;
  int mB = wave * 16;
  v8f acc0 = {0, 0, 0, 0, 0, 0, 0, 0};
  v8f acc1 = acc0, acc2 = acc0, acc3 = acc0;
  for (int k = 0; k < CDIM; k += 4) {
    int k0 = k + 2 * half;
    v2f a;
    a.x = wq[(mB + lm) * CDIM + k0];
    a.y = wq[(mB + lm) * CDIM + k0 + 1];
    v2f b0, b1, b2, b3;
    b0.x = lx[(0 + lm) * 196 + k0];
    b0.y = lx[(0 + lm) * 196 + k0 + 1];
    b1.x = lx[(16 + lm) * 196 + k0];
    b1.y = lx[(16 + lm) * 196 + k0 + 1];
    b2.x = lx[(32 + lm) * 196 + k0];
    b2.y = lx[(32 + lm) * 196 + k0 + 1];
    b3.x = lx[(48 + lm) * 196 + k0];
    b3.y = lx[(48 + lm) * 196 + k0 + 1];
    acc0 = wmma_f32(a, b0, acc0);
    acc1 = wmma_f32(a, b1, acc1);
    acc2 = wmma_f32(a, b2, acc2);
    acc3 = wmma_f32(a, b3, acc3);
  }
  for (int r = 0; r < 8; ++r) {
    int crow = mB + r + 8 * half;
    float bias = bq[crow];
    qb[crow * NTOK + 0 + lm] = acc0[r] + bias;
    qb[crow * NTOK + 16 + lm] = acc1[r] + bias;
    qb[crow * NTOK + 32 + lm] = acc2[r] + bias;
    qb[crow * NTOK + 48 + lm] = acc3[r] + bias;
  }
}

// -------------------------------------------- offset network + grid-sample xs
__global__ __launch_bounds__(256) void k_offset(const float* __restrict__ xin,
                                                const float* __restrict__ q,
                                                const float* __restrict__ dww,
                                                const float* __restrict__ dwb,
                                                const float* __restrict__ lng,
                                                const float* __restrict__ lnb,
                                                const float* __restrict__ pww,
                                                float* __restrict__ xs,
                                                float* __restrict__ pos_out,
                                                float* __restrict__ ref_out) {
  int bg = blockIdx.x;
  int b = bg / NGRP, g = bg % NGRP;
  __shared__ float so[16 * 68];  // [p][c]
  __shared__ float smu[16], srs[16];
  __shared__ float spos[32];
  const float* qg = q + ((size_t)b * CDIM + g * GCH) * NTOK;
  int tid = threadIdx.x;
  // depthwise 3x3 stride-2 conv (pad 1) on (64, 8, 8) -> (64, 4, 4)
  for (int e = tid; e < 1024; e += 256) {
    int p = e >> 6, c = e & 63;
    int py = p >> 2, px = p & 3;
    float s = dwb[c];
    for (int dy = 0; dy < 3; ++dy) {
      int yy = 2 * py + dy - 1;
      if (yy < 0 || yy > 7) continue;
      for (int dx = 0; dx < 3; ++dx) {
        int xx = 2 * px + dx - 1;
        if (xx < 0 || xx > 7) continue;
        s += dww[c * 9 + dy * 3 + dx] * qg[c * NTOK + yy * 8 + xx];
      }
    }
    so[p * 68 + c] = s;
  }
  __syncthreads();
  if (tid < 16) {  // channel LayerNorm stats per position
    float m = 0.0f;
    for (int c = 0; c < 64; ++c) m += so[tid * 68 + c];
    m *= (1.0f / 64.0f);
    float v = 0.0f;
    for (int c = 0; c < 64; ++c) { float d = so[tid * 68 + c] - m; v += d * d; }
    v *= (1.0f / 64.0f);
    smu[tid] = m;
    srs[tid] = rsqrtf(v + 1e-5f);
  }
  __syncthreads();
  for (int e = tid; e < 1024; e += 256) {  // LN + exact GELU
    int p = e >> 6, c = e & 63;
    float vl = (so[p * 68 + c] - smu[p]) * srs[p] * lng[c] + lnb[c];
    vl = 0.5f * vl * (1.0f + erff(vl * 0.70710678118654752f));
    so[p * 68 + c] = vl;
  }
  __syncthreads();
  if (tid < 32) {  // 1x1 conv to 2 channels, tanh, + reference points
    int p = tid >> 1, ch = tid & 1;
    float s = 0.0f;
    for (int c = 0; c < 64; ++c) s += pww[ch * 64 + c] * so[p * 68 + c];
    float off = tanhf(s) * (2.0f / 3.0f);  // tanh * 1/(Hk-1) * ORF
    int pc = ch ? (p & 3) : (p >> 2);      // ch0=y, ch1=x
    float ref = ((float)pc + 0.5f) * 0.5f - 1.0f;
    float po = off + ref;
    spos[p * 2 + ch] = po;
    size_t oidx = (((size_t)b * NGRP + g) * NSAMP + p) * 2 + ch;
    pos_out[oidx] = po;
    ref_out[oidx] = ref;
  }
  __syncthreads();
  // bilinear grid-sample of x (align_corners=True, zero pad)
  const float* xw = xin + (size_t)b * NTOK * CDIM + g * GCH;
  for (int e = tid; e < 1024; e += 256) {
    int p = e >> 6, c = e & 63;
    float gy = (spos[p * 2 + 0] + 1.0f) * 3.5f;
    float gx = (spos[p * 2 + 1] + 1.0f) * 3.5f;
    float x0 = floorf(gx), y0 = floorf(gy);
    float wx1 = gx - x0, wy1 = gy - y0;
    float acc = 0.0f;
    for (int jj = 0; jj < 2; ++jj)
      for (int ii = 0; ii < 2; ++ii) {
        float fx = x0 + (float)ii, fy = y0 + (float)jj;
        float w = (ii ? wx1 : 1.0f - wx1) * (jj ? wy1 : 1.0f - wy1);
        if (fx >= 0.0f && fx <= 7.0f && fy >= 0.0f && fy <= 7.0f) {
          int ix = (int)fx, iy = (int)fy;
          acc += w * xw[(size_t)(iy * 8 + ix) * CDIM + c];
        }
      }
    xs[((size_t)b * CDIM + g * GCH + c) * NSAMP + p] = acc;
  }
}

// ---------------------------------------------------------------- k/v project
// 12 waves: wave owns one M tile; K and V accumulators share the B operand.
__global__ __launch_bounds__(384) void k_kvproj(const float* __restrict__ xs,
                                                const float* __restrict__ wk,
                                                const float* __restrict__ bk,
                                                const float* __restrict__ wv,
                                                const float* __restrict__ bv,
                                                float* __restrict__ kb,
                                                float* __restrict__ vb) {
  __shared__ float lx[CDIM * NSAMP];  // [c][n], stride 16 (halves 32 banks apart)
  int b = blockIdx.x;
  const float* xw = xs + (size_t)b * CDIM * NSAMP;
  for (int i = threadIdx.x; i < CDIM * NSAMP; i += 384) lx[i] = xw[i];
  __syncthreads();
  int wave = threadIdx.x >> 5, lane = threadIdx.x & 31;
  int half = lane >> 4, lm = lane & 15;
  int mB = wave * 16;
  v8f accK = {0, 0, 0, 0, 0, 0, 0, 0};
  v8f accV = accK;
  for (int k = 0; k < CDIM; k += 4) {
    int k0 = k + 2 * half;
    v2f bm;
    bm.x = lx[k0 * NSAMP + lm];
    bm.y = lx[(k0 + 1) * NSAMP + lm];
    v2f aK, aV;
    aK.x = wk[(mB + lm) * CDIM + k0];
    aK.y = wk[(mB + lm) * CDIM + k0 + 1];
    aV.x = wv[(mB + lm) * CDIM + k0];
    aV.y = wv[(mB + lm) * CDIM + k0 + 1];
    accK = wmma_f32(aK, bm, accK);
    accV = wmma_f32(aV, bm, accV);
  }
  for (int r = 0; r < 8; ++r) {
    int crow = mB + r + 8 * half;
    kb[((size_t)b * CDIM + crow) * NSAMP + lm] = accK[r] + bk[crow];
    vb[((size_t)b * CDIM + crow) * NSAMP + lm] = accV[r] + bv[crow];
  }
}

// ------------------------------------------- attention: qk + bias + softmax + pv
__global__ __launch_bounds__(128) void k_attn(const float* __restrict__ q,
                                              const float* __restrict__ kb,
                                              const float* __restrict__ vb,
                                              const float* __restrict__ rpe,
                                              const float* __restrict__ pos,
                                              float* __restrict__ ao) {
  int bh = blockIdx.x;
  int b = bh / NHEAD, h = bh % NHEAD, g = h >> 1;
  __shared__ float srpe[225];           // 15x15 rpe slice for this head
  __shared__ float spos[32];            // 16 x (y,x)
  __shared__ float sattn[4 * 16 * 17];  // per-wave softmaxed tile, stride 17
  int tid = threadIdx.x;
  for (int i = tid; i < 225; i += 128) srpe[i] = rpe[h * 225 + i];
  if (tid < 32) spos[tid] = pos[(((size_t)b * NGRP + g) * NSAMP) * 2 + tid];
  __syncthreads();
  int wave = tid >> 5, lane = tid & 31, half = lane >> 4, lm = lane & 15;
  int mB = wave * 16;  // 16 query rows per wave
  size_t cbase = (size_t)b * CDIM + h * HCH;
  // attn(64x16) = qh^T(64x32) @ kh(32x16)
  v8f acc = {0, 0, 0, 0, 0, 0, 0, 0};
  for (int k = 0; k < HCH; k += 4) {
    int k0 = k + 2 * half;
    v2f a, bm;
    a.x = q[(cbase + k0) * NTOK + mB + lm];
    a.y = q[(cbase + k0 + 1) * NTOK + mB + lm];
    bm.x = kb[(cbase + k0) * NSAMP + lm];
    bm.y = kb[(cbase + k0 + 1) * NSAMP + lm];
    acc = wmma_f32(a, bm, acc);
  }
  const float scale = 0.17677669529663687f;  // 32^-0.5
  float* sat = sattn + wave * 272;
  for (int r = 0; r < 8; ++r) {
    int ml = r + 8 * half;
    int m = mB + ml, n = lm;
    // rpe bias via bilinear sample at disp = (qg - pos) * 0.5 on 15x15 grid
    float qy = (float)(m >> 3) * 0.25f - 0.875f;
    float qx = (float)(m & 7) * 0.25f - 0.875f;
    float dy = (qy - spos[n * 2 + 0]) * 0.5f;
    float dx = (qx - spos[n * 2 + 1]) * 0.5f;
    float gy = (dy + 1.0f) * 7.0f;
    float gx = (dx + 1.0f) * 7.0f;
    float x0 = floorf(gx), y0 = floorf(gy);
    float wx1 = gx - x0, wy1 = gy - y0;
    float bias = 0.0f;
    for (int jj = 0; jj < 2; ++jj)
      for (int ii = 0; ii < 2; ++ii) {
        float fx = x0 + (float)ii, fy = y0 + (float)jj;
        float w = (ii ? wx1 : 1.0f - wx1) * (jj ? wy1 : 1.0f - wy1);
        if (fx >= 0.0f && fx <= 14.0f && fy >= 0.0f && fy <= 14.0f)
          bias += w * srpe[(int)fy * 15 + (int)fx];
      }
    float av = acc[r] * scale + bias;
    // softmax across the 16 samples held by this 16-lane half
    float mx = av;
    for (int msk = 1; msk < 16; msk <<= 1) mx = fmaxf(mx, __shfl_xor(mx, msk, 32));
    float e = expf(av - mx);
    float sm = e;
    for (int msk = 1; msk < 16; msk <<= 1) sm += __shfl_xor(sm, msk, 32);
    sat[ml * 17 + n] = e / sm;
  }
  __syncthreads();
  // out^T(32x64): D(c2, m) = V(c2, n) @ P^T(n, m)
  v8f o0 = {0, 0, 0, 0, 0, 0, 0, 0};
  v8f o1 = {0, 0, 0, 0, 0, 0, 0, 0};
  for (int k = 0; k < NSAMP; k += 4) {
    int k0 = k + 2 * half;
    v2f bm, a0, a1;
    bm.x = sat[lm * 17 + k0];
    bm.y = sat[lm * 17 + k0 + 1];
    a0.x = vb[(cbase + lm) * NSAMP + k0];
    a0.y = vb[(cbase + lm) * NSAMP + k0 + 1];
    a1.x = vb[(cbase + 16 + lm) * NSAMP + k0];
    a1.y = vb[(cbase + 16 + lm) * NSAMP + k0 + 1];
    o0 = wmma_f32(a0, bm, o0);
    o1 = wmma_f32(a1, bm, o1);
  }
  for (int r = 0; r < 8; ++r) {
    int c0 = r + 8 * half;
    ao[(cbase + c0) * NTOK + mB + lm] = o0[r];
    ao[(cbase + 16 + c0) * NTOK + mB + lm] = o1[r];
  }
}

// ---------------------------------------------------------------- o projection
// 12 waves: wave owns one M tile, 4 N tiles share the A (weight) operand.
__global__ __launch_bounds__(384) void k_oproj(const float* __restrict__ ao,
                                               const float* __restrict__ wo,
                                               const float* __restrict__ bo,
                                               float* __restrict__ y) {
  __shared__ float la[CDIM * 72];  // [c][n], stride 72 (rows 2 apart -> +16 banks)
  int b = blockIdx.x;
  const float* aw = ao + (size_t)b * CDIM * NTOK;
  for (int i = threadIdx.x; i < CDIM * NTOK; i += 384) {
    int c = i >> 6, n = i & 63;
    la[c * 72 + n] = aw[i];
  }
  __syncthreads();
  int wave = threadIdx.x >> 5, lane = threadIdx.x & 31;
  int half = lane >> 4, lm = lane & 15;
  int mB = wave * 16;
  v8f acc0 = {0, 0, 0, 0, 0, 0, 0, 0};
  v8f acc1 = acc0, acc2 = acc0, acc3 = acc0;
  for (int k = 0; k < CDIM; k += 4) {
    int k0 = k + 2 * half;
    v2f a;
    a.x = wo[(mB + lm) * CDIM + k0];
    a.y = wo[(mB + lm) * CDIM + k0 + 1];
    v2f b0, b1, b2, b3;
    b0.x = la[k0 * 72 + 0 + lm];
    b0.y = la[(k0 + 1) * 72 + 0 + lm];
    b1.x = la[k0 * 72 + 16 + lm];
    b1.y = la[(k0 + 1) * 72 + 16 + lm];
    b2.x = la[k0 * 72 + 32 + lm];
    b2.y = la[(k0 + 1) * 72 + 32 + lm];
    b3.x = la[k0 * 72 + 48 + lm];
    b3.y = la[(k0 + 1) * 72 + 48 + lm];
    acc0 = wmma_f32(a, b0, acc0);
    acc1 = wmma_f32(a, b1, acc1);
    acc2 = wmma_f32(a, b2, acc2);
    acc3 = wmma_f32(a, b3, acc3);
  }
  for (int r = 0; r < 8; ++r) {
    int crow = mB + r + 8 * half;
    float bias = bo[crow];
    float* yb = y + ((size_t)b * CDIM + crow) * NTOK;
    yb[0 + lm] = acc0[r] + bias;
    yb[16 + lm] = acc1[r] + bias;
    yb[32 + lm] = acc2[r] + bias;
    yb[48 + lm] = acc3[r] + bias;
  }
}

extern "C" void kernel_launch(void* const* d_in, const int* in_sizes, int n_in,
                              void* d_out, int out_size, void* d_ws, size_t ws_size,
                              hipStream_t stream) {
  const float* x = (const float*)d_in[0];
  const float* wq = (const float*)d_in[1];
  const float* bq = (const float*)d_in[2];
  const float* wk = (const float*)d_in[3];
  const float* bk = (const float*)d_in[4];
  const float* wv = (const float*)d_in[5];
  const float* bv = (const float*)d_in[6];
  const float* wo = (const float*)d_in[7];
  const float* bo = (const float*)d_in[8];
  const float* off_dw_w = (const float*)d_in[9];
  const float* off_dw_b = (const float*)d_in[10];
  const float* ln_g = (const float*)d_in[11];
  const float* ln_b = (const float*)d_in[12];
  const float* off_pw_w = (const float*)d_in[13];
  const float* rpe = (const float*)d_in[14];

  int Bn = in_sizes[0] / (NTOK * CDIM);
  size_t QN = (size_t)Bn * CDIM * NTOK;   // q / attn-out size
  size_t XN = (size_t)Bn * CDIM * NSAMP;  // xs / k / v size
  float* ws = (float*)d_ws;
  float* qbuf = ws;
  float* xsbuf = qbuf + QN;
  float* kbuf = xsbuf + XN;
  float* vbuf = kbuf + XN;
  float* aobuf = vbuf + XN;

  float* out = (float*)d_out;
  float* pos_out = out + QN;  // (B, 3, 4, 4, 2)
  float* ref_out = pos_out + (size_t)Bn * NGRP * NSAMP * 2;

  k_qproj<<<dim3(Bn), dim3(384), 0, stream>>>(x, wq, bq, qbuf);
  k_offset<<<dim3(Bn * NGRP), dim3(256), 0, stream>>>(
      x, qbuf, off_dw_w, off_dw_b, ln_g, ln_b, off_pw_w, xsbuf, pos_out, ref_out);
  k_kvproj<<<dim3(Bn), dim3(384), 0, stream>>>(xsbuf, wk, bk, wv, bv, kbuf, vbuf);
  k_attn<<<dim3(Bn * NHEAD), dim3(128), 0, stream>>>(qbuf, kbuf, vbuf, rpe, pos_out, aobuf);
  k_oproj<<<dim3(Bn), dim3(384), 0, stream>>>(aobuf, wo, bo, out);
}